// Get_Cat_Feat_Tgt_82669530513430
// MI455X (gfx1250) — compile-verified
//
#include <hip/hip_runtime.h>

typedef float v2f __attribute__((ext_vector_type(2)));
typedef float v8f __attribute__((ext_vector_type(8)));

#define B_    4
#define KT_   128
#define C_    10
#define N_    16384
#define F_    32
#define M_    (KT_ * C_)      // 1280 queries per batch
#define KNN_  32
#define OUTC  (3 + F_)        // 35 output channels
#define NW_   4               // waves per workgroup (N split 4 ways)
#define NSPL  (N_ / NW_)      // 4096 points per wave

// One workgroup = 4 wave32 per (batch, 16-query tile); each wave sweeps N/4
// targets. Distances via V_WMMA_F32_16X16X4_F32 with the squared-norm folded
// into the K=3 slot:
//   A row (point)  = ( x,    y,    z,   ||p||^2 )
//   B col (query)  = (-2qx, -2qy, -2qz, 1       )
//   acc[r]         = ||p||^2 - 2 p.q   =>  d^2 = q^2 + acc[r]
// D layout: lane L, acc[r] = element (M = r + 8*(L>=16), N = L&15).
// Top-k runs on d^2; sqrt only on the final 32 per query. Final selection is
// an 8-way merge of the 8 sorted half-lists (4 waves x 2 lane halves).
__global__ __launch_bounds__(32 * NW_)
void knn_group_kernel(const float* __restrict__ cand,   // (B, M, 3)
                      const float* __restrict__ tgt,    // (B, N, 3)
                      const float* __restrict__ feat,   // (B, N, F)
                      float* __restrict__ out)          // (B, M, KNN, 35)
{
    __shared__ float sd[32 * NW_][KNN_ + 1];   // per-thread sorted top-32 d^2 (+pad col)
    __shared__ int   si[32 * NW_][KNN_ + 1];   // per-thread top-32 indices
    __shared__ int   fi[16][KNN_];             // merged per-query indices
    __shared__ float fw[16][KNN_];             // per-query dist, then normalized weights

    const int tid  = threadIdx.x;              // 0..127
    const int wave = tid >> 5;
    const int lane = tid & 31;
    const int tile = blockIdx.x % (M_ / 16);
    const int b    = blockIdx.x / (M_ / 16);
    const int j    = lane & 15;                // query slot / point slot within tile
    const int hi   = lane >> 4;                // 0: K=0,1 half, 1: K=2,3 half

    // ---- load my query (column j of the B matrix) ----
    const int m = tile * 16 + j;
    const float* qp = cand + ((size_t)b * M_ + m) * 3;
    const float qx = qp[0], qy = qp[1], qz = qp[2];
    const float q2 = qx * qx + qy * qy + qz * qz;

    v2f bmat;
    bmat.x = hi ? (-2.0f * qz) : (-2.0f * qx);
    bmat.y = hi ? 1.0f         : (-2.0f * qy);

    // ---- init per-thread top-k lists ----
    for (int k = 0; k < KNN_; ++k) { sd[tid][k] = 3.0e38f; si[tid][k] = 0; }
    float worst = 3.0e38f;

    const float* tb = tgt + (size_t)b * N_ * 3;
    const int pbeg = wave * NSPL;
    const int pend = pbeg + NSPL;

    // software pipeline: keep current chunk's point in registers, prefetch next
    const float* tp = tb + (size_t)(pbeg + j) * 3;
    float cx = tp[0], cy = tp[1], cz = tp[2];

    for (int p0 = pbeg; p0 < pend; p0 += 16) {
        const float* tn = (p0 + 16 < pend) ? (tp + 48) : tp;  // uniform clamp
        const float nx = tn[0], ny = tn[1], nz = tn[2];       // issue next load early

        const float r2own = cx * cx + cy * cy + cz * cz;      // ||point p0+j||^2
        v2f amat;
        amat.x = hi ? cz    : cx;
        amat.y = hi ? r2own : cy;

        v8f acc = {};
        acc = __builtin_amdgcn_wmma_f32_16x16x4_f32(
            /*neg_a=*/false, amat, /*neg_b=*/false, bmat,
            /*c_mod=*/(short)0, acc, /*reuse_a=*/false, /*reuse_b=*/false);

        #pragma unroll
        for (int r = 0; r < 8; ++r) {
            const int   pidx = p0 + r + 8 * hi;
            const float d2   = fmaxf(q2 + acc[r], 0.0f);  // clamp like reference
            if (d2 < worst) {                  // rare: insert into sorted LDS list
                const float old30 = sd[tid][KNN_ - 2];
                int pos = KNN_ - 1;
                #pragma clang loop unroll(disable)
                while (pos > 0 && sd[tid][pos - 1] > d2) {
                    sd[tid][pos] = sd[tid][pos - 1];
                    si[tid][pos] = si[tid][pos - 1];
                    --pos;
                }
                sd[tid][pos] = d2;
                si[tid][pos] = pidx;
                worst = fmaxf(old30, d2);      // new sd[31] without LDS reload
            }
        }
        cx = nx; cy = ny; cz = nz; tp = tn;
    }
    __syncthreads();

    // ---- 8-way merge of sorted half-lists -> exact sorted top-32 per query.
    // List l (0..7): wave l>>1, lane-half l&1 -> LDS row (l>>1)*32 + (l&1)*16 + j.
    // Thread j (< 16) merges for query j. heads/ptrs stay in registers
    // (constant indices in fully unrolled loops). Strict '<' prefers lower l
    // (= lower point index range) on ties. Sum of ptrs == k, so reads beyond
    // column 31 only touch the pad column on the very last advance (unused).
    if (tid < 16) {
        float head[8];
        int   ptr[8];
        #pragma unroll
        for (int l = 0; l < 8; ++l) {
            ptr[l]  = 0;
            head[l] = sd[(l >> 1) * 32 + (l & 1) * 16 + j][0];
        }
        float sum = 0.0f;
        #pragma clang loop unroll(disable)
        for (int k = 0; k < KNN_; ++k) {
            float bv = head[0]; int bl = 0; int bp = ptr[0];
            #pragma unroll
            for (int l = 1; l < 8; ++l)
                if (head[l] < bv) { bv = head[l]; bl = l; bp = ptr[l]; }
            fi[j][k] = si[(bl >> 1) * 32 + (bl & 1) * 16 + j][bp];
            #pragma unroll
            for (int l = 0; l < 8; ++l)
                if (l == bl) {
                    ptr[l]  = bp + 1;
                    head[l] = sd[(l >> 1) * 32 + (l & 1) * 16 + j][bp + 1];
                }
            const float d = sqrtf(bv);         // 32 sqrts per query, off hot path
            fw[j][k] = d;
            sum += d;
        }
        const float inv = 1.0f / sum;
        #pragma clang loop unroll(disable)
        for (int k = 0; k < KNN_; ++k) fw[j][k] *= inv;
    }
    __syncthreads();

    // ---- emit output: 4 queries per wave, lane n = neighbor n ----
    const float* fb = feat + (size_t)b * N_ * F_;
    for (int jj = wave * 4; jj < wave * 4 + 4; ++jj) {
        const int mq = tile * 16 + jj;
        const int gi = fi[jj][lane];
        const float* tp2 = tb + (size_t)gi * 3;
        const float* qq  = cand + ((size_t)b * M_ + mq) * 3;
        float* op = out + (((size_t)b * M_ + mq) * KNN_ + lane) * OUTC;
        op[0] = tp2[0] - qq[0];
        op[1] = tp2[1] - qq[1];
        op[2] = tp2[2] - qq[2];
        const float* fp = fb + (size_t)gi * F_;
        #pragma unroll
        for (int f = 0; f < F_; ++f)
            op[3 + f] = fp[f] * fw[jj][f];   // weight varies over FEATURE axis (ref quirk)
    }
}

extern "C" void kernel_launch(void* const* d_in, const int* in_sizes, int n_in,
                              void* d_out, int out_size, void* d_ws, size_t ws_size,
                              hipStream_t stream) {
    const float* cand = (const float*)d_in[0];   // candidate_pts (B,K,C,3) == (B,M,3)
    // d_in[1] = src_keypts — unused by the reference computation
    const float* tgt  = (const float*)d_in[2];   // tgt_pts_xyz (B,N,3)
    const float* feat = (const float*)d_in[3];   // tgt_deep_feat_pts (B,N,F)
    float* out = (float*)d_out;                  // (B,K,C,KNN,35) float32

    dim3 grid(B_ * (M_ / 16));                   // 320 blocks x 4 wave32
    knn_group_kernel<<<grid, 32 * NW_, 0, stream>>>(cand, tgt, feat, out);
}